// SinkhornDistance_23278722744477
// MI455X (gfx1250) — compile-verified
//
#include <hip/hip_runtime.h>
#include <math.h>

typedef __attribute__((ext_vector_type(2))) float v2f;
typedef __attribute__((ext_vector_type(8))) float v8f;

#define BATCH 16
#define NPTS 512
#define DDIM 128
#define EPSR 0.1f
#define INV_EPS 10.0f
#define NITER 20

// log(1/512 + 1e-8), constant-folded
__device__ __forceinline__ float log_marg() { return logf(1.0f / 512.0f + 1e-8f); }

// ---------------------------------------------------------------------------
// init: zero u, v, err[20], done flag, cost[16]
__global__ void sk_init(float* u, float* v, float* err, int* done, float* cost) {
    int idx = blockIdx.x * 256 + threadIdx.x;
    if (idx < BATCH * NPTS) { u[idx] = 0.0f; v[idx] = 0.0f; }
    if (idx < NITER) err[idx] = 0.0f;
    if (idx == NITER) *done = 0;
    if (idx < BATCH) cost[idx] = 0.0f;
}

// ---------------------------------------------------------------------------
// squared row norms: one wave (32 lanes) per row of 128 floats
__global__ void sk_norms(const float* __restrict__ X, float* __restrict__ out) {
    int row  = (blockIdx.x * blockDim.x + threadIdx.x) >> 5;   // 8192 rows
    int lane = threadIdx.x & 31;
    const float4* p = (const float4*)(X + (size_t)row * DDIM);
    float4 a = p[lane];
    float s = a.x * a.x + a.y * a.y + a.z * a.z + a.w * a.w;
    #pragma unroll
    for (int off = 16; off; off >>= 1) s += __shfl_xor(s, off, 32);
    if (lane == 0) out[row] = s;
}

// ---------------------------------------------------------------------------
// C[b,i,j] = xsq_i + ysq_j - 2 * (X Y^T)  via V_WMMA_F32_16X16X4_F32
// one wave per 16x16 tile; tiles = 16 * 32 * 32 = 16384; grid covers exactly.
__global__ void sk_cost_wmma(const float* __restrict__ X, const float* __restrict__ Y,
                             const float* __restrict__ xsq, const float* __restrict__ ysq,
                             float* __restrict__ C) {
    int tile = (blockIdx.x * blockDim.x + threadIdx.x) >> 5;
    int lane = threadIdx.x & 31;
    int b    = tile >> 10;          // 1024 tiles per batch
    int trow = (tile >> 5) & 31;
    int tcol = tile & 31;

    const float* Xb = X + ((size_t)b << 16);   // 512*128
    const float* Yb = Y + ((size_t)b << 16);

    // A frag 16x4 f32: lane l -> M = l&15, K = koff + {0,1}, koff = 2*(l>>4)
    // B frag 4x16 f32 (from Y^T): symmetric layout -> same float2 access on Y
    int mA   = lane & 15;
    int koff = (lane >> 4) << 1;
    const float* arow = Xb + (size_t)(trow * 16 + mA) * DDIM + koff;
    const float* brow = Yb + (size_t)(tcol * 16 + mA) * DDIM + koff;

    v8f acc = {0.f, 0.f, 0.f, 0.f, 0.f, 0.f, 0.f, 0.f};
    #pragma unroll
    for (int k = 0; k < DDIM; k += 4) {
        v2f a  = *(const v2f*)(arow + k);
        v2f bb = *(const v2f*)(brow + k);
        acc = __builtin_amdgcn_wmma_f32_16x16x4_f32(
            false, a, false, bb, (short)0, acc, false, false);
    }

    // D layout: acc[g] is element (M = g + 8*(lane>>4), N = lane&15)
    int n     = lane & 15;
    int mhalf = (lane >> 4) << 3;
    float ys  = ysq[b * NPTS + tcol * 16 + n];
    float* Cb = C + ((size_t)b << 18) + (size_t)(trow * 16) * NPTS + tcol * 16 + n;
    #pragma unroll
    for (int g = 0; g < 8; ++g) {
        int mm = mhalf + g;
        float xs = xsq[b * NPTS + trow * 16 + mm];
        Cb[(size_t)mm * NPTS] = xs + ys - 2.0f * acc[g];
    }
}

// ---------------------------------------------------------------------------
// u-update: one wave per (b,i) row; logsumexp over j (512), coalesced float4
__global__ void sk_u_update(const float* __restrict__ C, const float* __restrict__ v,
                            float* __restrict__ u, float* __restrict__ err_t,
                            const int* __restrict__ done) {
    if (*done) return;
    int row  = blockIdx.x * 8 + (threadIdx.x >> 5);   // 8192 rows
    int lane = threadIdx.x & 31;
    int b    = row >> 9;
    const float4* Crow = (const float4*)(C + ((size_t)row << 9));
    const float4* vb   = (const float4*)(v + ((size_t)b << 9));
    float u_i = u[row];

    float t[16];
    float m = -INFINITY;
    #pragma unroll
    for (int k = 0; k < 4; ++k) {
        float4 c4 = Crow[lane + 32 * k];
        float4 v4 = vb[lane + 32 * k];
        int e = 4 * k;
        t[e + 0] = (u_i + v4.x - c4.x) * INV_EPS;
        t[e + 1] = (u_i + v4.y - c4.y) * INV_EPS;
        t[e + 2] = (u_i + v4.z - c4.z) * INV_EPS;
        t[e + 3] = (u_i + v4.w - c4.w) * INV_EPS;
    }
    #pragma unroll
    for (int e = 0; e < 16; ++e) m = fmaxf(m, t[e]);
    #pragma unroll
    for (int off = 16; off; off >>= 1) m = fmaxf(m, __shfl_xor(m, off, 32));
    float s = 0.0f;
    #pragma unroll
    for (int e = 0; e < 16; ++e) s += expf(t[e] - m);
    #pragma unroll
    for (int off = 16; off; off >>= 1) s += __shfl_xor(s, off, 32);

    float lse   = m + logf(s);
    float u_new = EPSR * (log_marg() - lse) + u_i;
    if (lane == 0) {
        u[row] = u_new;
        atomicAdd(err_t, fabsf(u_new - u_i));
    }
}

// ---------------------------------------------------------------------------
// v-update: block = 256 threads handles 64 columns of one batch, 4-way i-split
// online logsumexp over i with coalesced row-slice reads; combine via LDS
__global__ void sk_v_update(const float* __restrict__ C, const float* __restrict__ u,
                            float* __restrict__ v, const int* __restrict__ done) {
    if (*done) return;
    int tid  = threadIdx.x;
    int b    = blockIdx.x >> 3;           // 8 blocks per batch
    int jb   = (blockIdx.x & 7) * 64;
    int col  = jb + (tid & 63);
    int part = tid >> 6;                  // 0..3 split over i

    __shared__ float su[NPTS];
    su[tid]       = u[(b << 9) + tid];
    su[tid + 256] = u[(b << 9) + tid + 256];
    __syncthreads();

    const float* Cb = C + ((size_t)b << 18);
    float vj = v[(b << 9) + col];

    float m = -INFINITY, s = 0.0f;
    for (int kk = 0; kk < 128; ++kk) {
        int i = part + 4 * kk;
        float val = (su[i] + vj - Cb[(size_t)i * NPTS + col]) * INV_EPS;
        if (val <= m) {
            s += expf(val - m);
        } else {
            s = s * expf(m - val) + 1.0f;
            m = val;
        }
    }

    __shared__ float sm[256], ss[256];
    sm[tid] = m; ss[tid] = s;
    __syncthreads();
    if (tid < 64) {
        float M = sm[tid];
        #pragma unroll
        for (int p = 1; p < 4; ++p) M = fmaxf(M, sm[tid + 64 * p]);
        float S = 0.0f;
        #pragma unroll
        for (int p = 0; p < 4; ++p) S += ss[tid + 64 * p] * expf(sm[tid + 64 * p] - M);
        float lse = M + logf(S);
        v[(b << 9) + jb + tid] = EPSR * (log_marg() - lse) + vj;
    }
}

// ---------------------------------------------------------------------------
// gate: done |= (err/B < 0.1)
__global__ void sk_done_gate(const float* __restrict__ err_t, int* done) {
    if (threadIdx.x == 0 && blockIdx.x == 0) {
        if ((*err_t) * (1.0f / (float)BATCH) < 0.1f) *done = 1;
    }
}

// ---------------------------------------------------------------------------
// pi = exp((-C+u+v)/eps); cost[b] = sum(pi*C). 1024 blocks x 256 thr, 16 f/thr
__global__ void sk_pi_cost(const float* __restrict__ C, const float* __restrict__ u,
                           const float* __restrict__ v, float* __restrict__ pi,
                           float* __restrict__ cost) {
    int tid = threadIdx.x;
    int b   = blockIdx.x >> 6;            // 64 blocks per batch
    const float4* C4 = (const float4*)C;
    float4*       P4 = (float4*)pi;
    const float* ub = u + (b << 9);
    const float* vb = v + (b << 9);

    size_t q = (size_t)blockIdx.x * 1024 + tid;   // float4 index
    float partial = 0.0f;
    #pragma unroll
    for (int k = 0; k < 4; ++k, q += 256) {
        float4 c4 = C4[q];
        size_t e  = q << 2;
        int i = (int)((e >> 9) & 511);
        int j = (int)(e & 511);
        float ui  = ub[i];
        float4 v4 = *(const float4*)(vb + j);
        float4 p;
        p.x = expf((ui + v4.x - c4.x) * INV_EPS);
        p.y = expf((ui + v4.y - c4.y) * INV_EPS);
        p.z = expf((ui + v4.z - c4.z) * INV_EPS);
        p.w = expf((ui + v4.w - c4.w) * INV_EPS);
        partial += p.x * c4.x + p.y * c4.y + p.z * c4.z + p.w * c4.w;
        P4[q] = p;
    }

    __shared__ float red[256];
    red[tid] = partial;
    __syncthreads();
    for (int s = 128; s > 0; s >>= 1) {
        if (tid < s) red[tid] += red[tid + s];
        __syncthreads();
    }
    if (tid == 0) atomicAdd(cost + b, red[0]);
}

// ---------------------------------------------------------------------------
extern "C" void kernel_launch(void* const* d_in, const int* in_sizes, int n_in,
                              void* d_out, int out_size, void* d_ws, size_t ws_size,
                              hipStream_t stream) {
    const float* x = (const float*)d_in[0];   // [16,512,128]
    const float* y = (const float*)d_in[1];   // [16,512,128]

    float* out  = (float*)d_out;
    float* cost = out;                                     // [16]
    float* pi   = out + BATCH;                             // [16,512,512]
    float* C    = out + BATCH + (size_t)BATCH * NPTS * NPTS;

    float* ws  = (float*)d_ws;
    float* xsq = ws;                    // 8192
    float* ysq = ws + 8192;             // 8192
    float* u   = ws + 16384;            // 8192
    float* v   = ws + 24576;            // 8192
    float* err = ws + 32768;            // 20
    int*  done = (int*)(ws + 32800);

    sk_init<<<32, 256, 0, stream>>>(u, v, err, done, cost);
    sk_norms<<<1024, 256, 0, stream>>>(x, xsq);
    sk_norms<<<1024, 256, 0, stream>>>(y, ysq);
    sk_cost_wmma<<<2048, 256, 0, stream>>>(x, y, xsq, ysq, C);

    for (int t = 0; t < NITER; ++t) {
        sk_u_update<<<1024, 256, 0, stream>>>(C, v, u, err + t, done);
        sk_v_update<<<128, 256, 0, stream>>>(C, u, v, done);
        sk_done_gate<<<1, 32, 0, stream>>>(err + t, done);
    }

    sk_pi_cost<<<1024, 256, 0, stream>>>(C, u, v, pi, cost);
}